// LLaDA2GroupLimitedRouter_40905268527495
// MI455X (gfx1250) — compile-verified
//
#include <hip/hip_runtime.h>

// LLaDA2 group-limited router (DeepSeek-style): TOP_K=8, 256 experts,
// 8 groups x 32 experts, keep top-4 groups by (top-2 sum), sigmoid scoring,
// renormalize, scale=2.5.
//
// One wave32 per token; lane L owns experts [8L, 8L+8).
// v2: selection runs on packed sortable u64 keys (value<<32 | 255-idx), one
// per-lane sort + 8 wave-wide u64-max butterflies via ds_swizzle_b32.
// Winner weights are recovered as (biased_value - bias[idx]) at the end,
// so no weight field travels through any reduction.

#define TOPK    8
#define NEXP    256
#define SCALE_F 2.5f
#define NEG_INF (-__builtin_inff())

template <int XOR>
__device__ __forceinline__ unsigned swz_xor_u(unsigned x) {
  // group-of-32 swizzle: new_lane = (lane & 0x1f) ^ XOR
  return (unsigned)__builtin_amdgcn_ds_swizzle((int)x, (XOR << 10) | 0x1f);
}
template <int XOR>
__device__ __forceinline__ float swz_xor_f(float x) {
  return __int_as_float(
      __builtin_amdgcn_ds_swizzle(__float_as_int(x), (XOR << 10) | 0x1f));
}
template <int XOR>
__device__ __forceinline__ unsigned long long swz_xor_u64(unsigned long long x) {
  unsigned lo = swz_xor_u<XOR>((unsigned)x);
  unsigned hi = swz_xor_u<XOR>((unsigned)(x >> 32));
  return ((unsigned long long)hi << 32) | lo;
}

__device__ __forceinline__ float readlane_f(float x, int lane) {
  return __int_as_float(__builtin_amdgcn_readlane(__float_as_int(x), lane));
}

// Order-preserving f32 <-> u32 maps (exact round-trip).
__device__ __forceinline__ unsigned f2key(float x) {
  int b = __float_as_int(x);
  return (unsigned)(b ^ ((b >> 31) | 0x80000000));
}
__device__ __forceinline__ float key2f(unsigned k) {
  int b = (int)k;
  return __int_as_float(b ^ (((~b) >> 31) | 0x80000000));
}

__global__ __launch_bounds__(256) void
router_group_topk_kernel(const float* __restrict__ logits,
                         const float* __restrict__ bias,
                         float* __restrict__ out_w,
                         int* __restrict__ out_id,
                         int tokens) {
  const int lane  = threadIdx.x & 31;
  const int wave  = threadIdx.x >> 5;
  const int token = blockIdx.x * 8 + wave;
  if (token >= tokens) return;

  const float* row  = logits + (size_t)token * NEXP;
  const int    base = lane * 8;

  // ---- load 8 logits + 8 biases per lane (wave reads a contiguous 1KB) ----
  float4 r0 = *reinterpret_cast<const float4*>(row + base);
  float4 r1 = *reinterpret_cast<const float4*>(row + base + 4);
  float4 b0 = *reinterpret_cast<const float4*>(bias + base);
  float4 b1 = *reinterpret_cast<const float4*>(bias + base + 4);

  float lg[8] = {r0.x, r0.y, r0.z, r0.w, r1.x, r1.y, r1.z, r1.w};
  float bi[8] = {b0.x, b0.y, b0.z, b0.w, b1.x, b1.y, b1.z, b1.w};

  float v[8];  // biased selection key: sigmoid(logit) + bias
#pragma unroll
  for (int i = 0; i < 8; ++i) {
    float s = 1.0f / (1.0f + expf(-lg[i]));
    v[i]    = s + bi[i];
  }

  // ---- group score = sum of top-2 biased scores in each 32-expert group ---
  float m1 = NEG_INF, m2 = NEG_INF;
#pragma unroll
  for (int i = 0; i < 8; ++i) {
    float x  = v[i];
    float hi = fmaxf(m1, x);
    float lo = fmaxf(fminf(m1, x), m2);
    m1 = hi;
    m2 = lo;
  }
  {  // merge across the 4 lanes of the group (lanes 4g..4g+3)
    float o1 = swz_xor_f<1>(m1), o2 = swz_xor_f<1>(m2);
    float hi = fmaxf(m1, o1);
    float lo = fmaxf(fminf(m1, o1), fmaxf(m2, o2));
    m1 = hi; m2 = lo;
  }
  {
    float o1 = swz_xor_f<2>(m1), o2 = swz_xor_f<2>(m2);
    float hi = fmaxf(m1, o1);
    float lo = fmaxf(fminf(m1, o1), fmaxf(m2, o2));
    m1 = hi; m2 = lo;
  }
  const float gscore = m1 + m2;

  // ---- broadcast 8 group scores (lane 4g holds group g) -> scalar top-4 ---
  float gs[8];
#pragma unroll
  for (int g = 0; g < 8; ++g) gs[g] = readlane_f(gscore, g * 4);

  unsigned gmask = 0u;
#pragma unroll
  for (int k = 0; k < 4; ++k) {
    float best = NEG_INF;
    int   bg   = 0;
#pragma unroll
    for (int g = 0; g < 8; ++g) {
      bool take = (((gmask >> g) & 1u) == 0u) && (gs[g] > best);
      best = take ? gs[g] : best;
      bg   = take ? g : bg;
    }
    gmask |= (1u << bg);
  }
  const bool keep = ((gmask >> (lane >> 2)) & 1u) != 0u;

  // ---- build sortable keys: (value_key << 32) | (255 - expert_idx) --------
  // Losing groups get hi=0 (below every real candidate, incl. -inf scores).
  unsigned long long kk[8];
#pragma unroll
  for (int i = 0; i < 8; ++i) {
    unsigned hi = keep ? f2key(v[i]) : 0u;
    kk[i] = ((unsigned long long)hi << 32) | (unsigned)(255 - (base + i));
  }

  // ---- per-lane descending sort (19-comparator optimal network) -----------
#define CE(i, j)                                                   \
  {                                                                \
    unsigned long long a = kk[i], b = kk[j];                       \
    bool sw = a < b;                                               \
    kk[i] = sw ? b : a;                                            \
    kk[j] = sw ? a : b;                                            \
  }
  CE(0, 1) CE(2, 3) CE(4, 5) CE(6, 7)
  CE(0, 2) CE(1, 3) CE(4, 6) CE(5, 7)
  CE(1, 2) CE(5, 6) CE(0, 4) CE(3, 7)
  CE(1, 5) CE(2, 6)
  CE(1, 4) CE(3, 6)
  CE(2, 4) CE(3, 5)
  CE(3, 4)
#undef CE

  // ---- 8 rounds: wave-wide u64 max over list heads, pop the winner --------
  unsigned my_hi = f2key(0.0f);  // lane t (t<8) records winner t here
  int      my_id = 0;

#pragma unroll
  for (int t = 0; t < TOPK; ++t) {
    unsigned long long w = kk[0];
#define RED(OFF)                                                   \
    {                                                              \
      unsigned long long o = swz_xor_u64<OFF>(w);                  \
      w = (o > w) ? o : w;                                         \
    }
    RED(1) RED(2) RED(4) RED(8) RED(16)
#undef RED
    const unsigned wlo = (unsigned)w;
    const int      idx = 255 - (int)wlo;
    if (lane == t) {
      my_hi = (unsigned)(w >> 32);
      my_id = idx;
    }
    // owner pops its head (keys are unique in the low word)
    const bool own = ((unsigned)kk[0] == wlo);
#pragma unroll
    for (int i = 0; i < 7; ++i) kk[i] = own ? kk[i + 1] : kk[i];
    kk[7] = own ? 0ull : kk[7];
  }

  // ---- recover weights: s[idx] = biased_value - bias[idx] ------------------
  const float val = key2f(my_hi);
  const float bfx = bias[my_id & 255];
  float       wgt = val - bfx;  // winners live on lanes 0..7 (lane t = slot t)

  // sum over lanes 0..7 (xor 1/2/4 stays inside the group of 8)
  float wsum = wgt;
  wsum += swz_xor_f<1>(wsum);
  wsum += swz_xor_f<2>(wsum);
  wsum += swz_xor_f<4>(wsum);

  if (lane < TOPK) {
    float w = wgt / (wsum + 1e-20f) * SCALE_F;
    out_w[(size_t)token * TOPK + lane]  = w;
    out_id[(size_t)token * TOPK + lane] = my_id;
  }
}

extern "C" void kernel_launch(void* const* d_in, const int* in_sizes, int n_in,
                              void* d_out, int out_size, void* d_ws,
                              size_t ws_size, hipStream_t stream) {
  const float* logits = (const float*)d_in[0];  // [T, 256] f32
  const float* bias   = (const float*)d_in[1];  // [256] f32
  const int tokens    = in_sizes[0] / NEXP;

  // d_out = [ topk_weights f32 (T*8) | topk_ids i32 (T*8) ]
  float* out_w  = (float*)d_out;
  int*   out_id = (int*)(out_w + (size_t)tokens * TOPK);

  const int waves_per_block = 8;  // 256 threads = 8 x wave32
  dim3 block(256);
  dim3 grid((tokens + waves_per_block - 1) / waves_per_block);
  router_group_topk_kernel<<<grid, block, 0, stream>>>(logits, bias, out_w,
                                                       out_id, tokens);
}